// GCNConv_SS_HH_90555090468954
// MI455X (gfx1250) — compile-verified
//
#include <hip/hip_runtime.h>

typedef float v2f __attribute__((ext_vector_type(2)));
typedef float v8f __attribute__((ext_vector_type(8)));

#define CH 128
#define LDS_PITCH 132   // 132 mod 64 = 4 -> conflict-free ds_load_b64 across 16 lanes

// ---------------------------------------------------------------------------
// Phase 1: x_t = x @ W^T + b   via V_WMMA_F32_16X16X4_F32 (fp32-exact path)
// One wave computes a 16x128 output tile (8 accumulators of 16x16).
// ---------------------------------------------------------------------------
__global__ __launch_bounds__(128) void gcn_gemm_wmma(
    const float* __restrict__ x, const float* __restrict__ W,
    const float* __restrict__ bias, float* __restrict__ xt, int ntiles) {
  __shared__ float wl[CH * LDS_PITCH];

  const int tid = threadIdx.x;
  // Cooperative, coalesced copy of W (128x128 row-major [out,in]) into padded LDS.
  for (int i = tid; i < CH * CH; i += 128) {
    const int r = i >> 7;
    const int c = i & (CH - 1);
    wl[r * LDS_PITCH + c] = W[i];
  }
  __syncthreads();

  const int wave = tid >> 5;
  const int lane = tid & 31;
  const int rt = blockIdx.x * 4 + wave;          // 16-row tile index
  if (rt >= ntiles) return;                      // wave-uniform: EXEC stays all-1s
  const int rb  = rt * 16;
  const int lh  = lane & 15;                     // matrix row/col within halfwave
  const int khi = lane >> 4;                     // 0: K={k,k+1}, 1: K={k+2,k+3}

  // Fold bias into initial C: C/D element (vgpr i, lane) -> col n = cb*16+lh.
  v8f acc[8];
#pragma unroll
  for (int cb = 0; cb < 8; ++cb) {
    const float bv = bias[cb * 16 + lh];
    acc[cb] = (v8f){bv, bv, bv, bv, bv, bv, bv, bv};
  }

  // A layout (32-bit A 16x4): lane lh holds x[rb+lh][k + 2*khi .. +1]  (8B load)
  const float* arow = &x[(size_t)(rb + lh) * CH + 2 * khi];

  for (int k = 0; k < CH; k += 4) {
    const v2f a = *(const v2f*)(arow + k);
#pragma unroll
    for (int cb = 0; cb < 8; ++cb) {
      // B = W^T tile: B[k'][n] = W[n][k'];  n = cb*16 + lh, mirrored K split.
      const v2f bm = *(const v2f*)&wl[(cb * 16 + lh) * LDS_PITCH + k + 2 * khi];
      acc[cb] = __builtin_amdgcn_wmma_f32_16x16x4_f32(
          false, a, false, bm, (short)0, acc[cb], false, false);
    }
  }

  // C/D layout: vgpr i, lanes 0-15 -> M=i, lanes 16-31 -> M=i+8; N = lane&15.
#pragma unroll
  for (int cb = 0; cb < 8; ++cb) {
#pragma unroll
    for (int i = 0; i < 8; ++i) {
      xt[(size_t)(rb + i + 8 * khi) * CH + cb * 16 + lh] = acc[cb][i];
    }
  }
}

// ---------------------------------------------------------------------------
// Phase 0: zero the accumulator (d_out) — float4 stores.
// ---------------------------------------------------------------------------
__global__ void gcn_zero(float4* __restrict__ p, int n4) {
  const int i = blockIdx.x * blockDim.x + threadIdx.x;
  if (i < n4) p[i] = make_float4(0.f, 0.f, 0.f, 0.f);
}

// ---------------------------------------------------------------------------
// Phase 2: scatter-add messages. One wave per edge; each lane gathers a
// float4 (512B coalesced per wave from L2-resident xt) and issues 4
// non-returning global_atomic_add_f32 into out[row].
// ---------------------------------------------------------------------------
__global__ __launch_bounds__(256) void gcn_scatter(
    const float* __restrict__ xt, const int* __restrict__ ei,
    float* __restrict__ out, int nedges) {
  const int e = blockIdx.x * (blockDim.x >> 5) + (threadIdx.x >> 5);
  if (e >= nedges) return;
  const int lane = threadIdx.x & 31;
  const int row = ei[e];            // edge_index[0][e]  (dst segment)
  const int col = ei[nedges + e];   // edge_index[1][e]  (src node)

  const float4 v = ((const float4*)&xt[(size_t)col * CH])[lane];
  float* dst = &out[(size_t)row * CH + lane * 4];
  atomicAdd(dst + 0, v.x);
  atomicAdd(dst + 1, v.y);
  atomicAdd(dst + 2, v.z);
  atomicAdd(dst + 3, v.w);
}

// ---------------------------------------------------------------------------
// Phase 3: out = tanh(out) in-place, float4 vectorized.
// ---------------------------------------------------------------------------
__global__ void gcn_tanh(float4* __restrict__ p, int n4) {
  const int i = blockIdx.x * blockDim.x + threadIdx.x;
  if (i < n4) {
    float4 v = p[i];
    v.x = tanhf(v.x);
    v.y = tanhf(v.y);
    v.z = tanhf(v.z);
    v.w = tanhf(v.w);
    p[i] = v;
  }
}

// ---------------------------------------------------------------------------
extern "C" void kernel_launch(void* const* d_in, const int* in_sizes, int n_in,
                              void* d_out, int out_size, void* d_ws, size_t ws_size,
                              hipStream_t stream) {
  const float* x    = (const float*)d_in[0];   // [N, 128] fp32
  const int*   ei   = (const int*)d_in[1];     // [2, E] int (jax int64 -> int32 under default x64-off)
  const float* W    = (const float*)d_in[2];   // [128, 128] fp32
  const float* bias = (const float*)d_in[3];   // [128] fp32

  float* out = (float*)d_out;                  // [N, 128] fp32
  float* xt  = (float*)d_ws;                   // scratch: [N, 128] fp32

  const int N = in_sizes[0] / CH;              // 50000
  const int E = in_sizes[1] / 2;               // 800000
  const int total = N * CH;
  const int n4 = total / 4;

  // Phase 0: zero accumulator
  gcn_zero<<<(n4 + 255) / 256, 256, 0, stream>>>((float4*)out, n4);

  // Phase 1: WMMA GEMM  xt = x @ W^T + b
  const int ntiles = N / 16;                   // 3125, exact
  gcn_gemm_wmma<<<(ntiles + 3) / 4, 128, 0, stream>>>(x, W, bias, xt, ntiles);

  // Phase 2: edge scatter-add (one wave per edge, 8 waves per block)
  gcn_scatter<<<(E + 7) / 8, 256, 0, stream>>>(xt, ei, out, E);

  // Phase 3: tanh epilogue
  gcn_tanh<<<(n4 + 255) / 256, 256, 0, stream>>>((float4*)out, n4);
}